// MotionR2BC_14250701488835
// MI455X (gfx1250) — compile-verified
//
#include <hip/hip_runtime.h>

// MotionR2BC: q,v,a = f(t, r0, theta0, omega0)  — elementwise, bandwidth-bound.
// Layout: t (B,T,1) row-major; outputs q|v|a each (B,T,2), concatenated in d_out.

typedef __attribute__((ext_vector_type(4))) float v4f;  // native clang vector: b128 + NT ok

#define TPB 256
#define VPT 4   // t-values per thread -> 1x b128 nt load, 6x b128 nt stores

__global__ __launch_bounds__(TPB) void motion_r2bc_kernel(
    const float* __restrict__ t,
    const float* __restrict__ r0,
    const float* __restrict__ theta0,
    const float* __restrict__ omega0,
    float* __restrict__ q,
    float* __restrict__ v,
    float* __restrict__ a,
    int T)
{
    const int b = blockIdx.y;

    // Wave-uniform row scalars -> s_load_b32 (scalar cache), no VGPR traffic.
    const float r   = r0[b];
    const float th0 = theta0[b];
    const float w   = omega0[b];
    const float rw   = r * w;
    const float rww  = rw * w;
    const float nrw  = -rw;
    const float nrww = -rww;

    const int tIdx = (blockIdx.x * TPB + (int)threadIdx.x) * VPT;
    if (tIdx >= T) return;

    const long long rowBase = (long long)b * (long long)T;

    if (tIdx + VPT <= T) {
        // ---- vector fast path: one b128 NT load of t ----
        const v4f tv = __builtin_nontemporal_load(
            (const v4f*)(t + rowBase + tIdx));

        // theta in revolutions: one shared mul feeds both v_sin_f32 and v_cos_f32.
        const float k = 0.15915494309189535f; // 1/(2*pi) — inline const on gfx1250
        float th[VPT] = { fmaf(w, tv.x, th0), fmaf(w, tv.y, th0),
                          fmaf(w, tv.z, th0), fmaf(w, tv.w, th0) };
        float s[VPT], c[VPT];
#pragma unroll
        for (int i = 0; i < VPT; ++i) {
            const float tn = th[i] * k;
            s[i] = __builtin_amdgcn_sinf(tn);   // v_sin_f32
            c[i] = __builtin_amdgcn_cosf(tn);   // v_cos_f32
        }

        const long long o = (rowBase + tIdx) * 2;  // inner dim 2

        v4f q0 = { r * c[0],  r * s[0],  r * c[1],  r * s[1] };
        v4f q1 = { r * c[2],  r * s[2],  r * c[3],  r * s[3] };
        v4f v0 = { nrw * s[0], rw * c[0], nrw * s[1], rw * c[1] };
        v4f v1 = { nrw * s[2], rw * c[2], nrw * s[3], rw * c[3] };
        v4f a0 = { nrww * c[0], nrww * s[0], nrww * c[1], nrww * s[1] };
        v4f a1 = { nrww * c[2], nrww * s[2], nrww * c[3], nrww * s[3] };

        // Write-once 384MB stream: NT so we don't rinse L2 (output is 2x L2 size).
        __builtin_nontemporal_store(q0, (v4f*)(q + o));
        __builtin_nontemporal_store(q1, (v4f*)(q + o + 4));
        __builtin_nontemporal_store(v0, (v4f*)(v + o));
        __builtin_nontemporal_store(v1, (v4f*)(v + o + 4));
        __builtin_nontemporal_store(a0, (v4f*)(a + o));
        __builtin_nontemporal_store(a1, (v4f*)(a + o + 4));
    } else {
        // ---- scalar tail (unused for T=16384, kept for generality) ----
        for (int i = tIdx; i < T; ++i) {
            const float theta = fmaf(w, t[rowBase + i], th0);
            const float tn = theta * 0.15915494309189535f;
            const float si = __builtin_amdgcn_sinf(tn);
            const float co = __builtin_amdgcn_cosf(tn);
            const long long o = (rowBase + i) * 2;
            q[o]     = r * co;    q[o + 1] = r * si;
            v[o]     = nrw * si;  v[o + 1] = rw * co;
            a[o]     = nrww * co; a[o + 1] = nrww * si;
        }
    }
}

extern "C" void kernel_launch(void* const* d_in, const int* in_sizes, int n_in,
                              void* d_out, int out_size, void* d_ws, size_t ws_size,
                              hipStream_t stream) {
    const float* t      = (const float*)d_in[0];  // (B,T,1)
    const float* r0     = (const float*)d_in[1];  // (B,1)
    const float* theta0 = (const float*)d_in[2];  // (B,1)
    const float* omega0 = (const float*)d_in[3];  // (B,1)

    const int B = in_sizes[1];              // r0 flat count == B
    const int T = in_sizes[0] / B;          // t flat count == B*T

    float* out = (float*)d_out;
    const size_t plane = (size_t)B * (size_t)T * 2;  // each of q,v,a
    float* q = out;
    float* v = out + plane;
    float* a = out + 2 * plane;

    dim3 block(TPB);
    dim3 grid((unsigned)((T + TPB * VPT - 1) / (TPB * VPT)), (unsigned)B);
    motion_r2bc_kernel<<<grid, block, 0, stream>>>(t, r0, theta0, omega0, q, v, a, T);
}